// UpsampleDepthOutputReLu_9543417332268
// MI455X (gfx1250) — compile-verified
//
#include <hip/hip_runtime.h>
#include <hip/hip_bf16.h>

// Shapes (compile-time constants from the reference):
//   x:      (4, 64, 96, 320)  fp32
//   weight: (1, 64, 1, 1)     fp32  -> w[c]
//   bias:   (1,)              fp32
//   out:    (4, 1, 384, 1280) fp32
//
// Rewrite: relu( up4(sum_c w_c * x_c) + bias )  — reduce channels first,
// then bilinear-upsample the tiny (4,96,320) tensor. Memory-bound:
// ~39 MB total traffic -> ~1.7 us at 23.3 TB/s HBM.

#define B_   4
#define C_   64
#define H_   96
#define W_   320
#define HW_  (H_ * W_)          // 30720
#define CHW_ (C_ * HW_)
#define HO_  384
#define WO_  1280
#define NOUT_ (B_ * HO_ * WO_)  // 1966080
#define NZ_   (B_ * HW_)        // 122880

typedef __attribute__((ext_vector_type(2))) float v2f;
typedef __attribute__((ext_vector_type(8))) float v8f;

// ---------------------------------------------------------------------------
// Kernel 1: z[b][h][w] = sum_c w[c] * x[b][c][h][w]   via V_WMMA_F32_16X16X4_F32
//
// Each wave owns 16 consecutive flat positions p = wave*16 .. +15 (never
// crosses a batch boundary since HW_ % 16 == 0). Per K-chunk of 4 channels:
//   A (16x4 f32, 2 VGPRs): lane l (m = l&15, half = l>>4):
//       VGPR0 = x[c0 + 2*half    ][p_m]   (K = 0 | 2)
//       VGPR1 = x[c0 + 2*half + 1][p_m]   (K = 1 | 3)
//   B (4x16 f32, 2 VGPRs): w-chunk broadcast along N:
//       VGPR0 = w[c0 + 2*half], VGPR1 = w[c0 + 2*half + 1]
// D[m][n] is then identical across n; lanes with N==0 (lanes 0 and 16) hold
// M = r and M = r+8 in acc[r], and write the 16 results.
// ---------------------------------------------------------------------------
__global__ __launch_bounds__(256) void
upsample_reduce_c_kernel(const float* __restrict__ x,
                         const float* __restrict__ w,
                         float* __restrict__ z) {
  const int tid  = blockIdx.x * 256 + threadIdx.x;
  const int wave = tid >> 5;
  const int lane = threadIdx.x & 31;
  const int m    = lane & 15;
  const int half = lane >> 4;

  const int pbase = wave * 16;          // 16 consecutive z positions
  const int b     = pbase / HW_;
  const int s     = pbase - b * HW_;

  const float* xb = x + (size_t)b * CHW_ + (size_t)(s + m);

  v8f acc = {};
#pragma unroll
  for (int c0 = 0; c0 < C_; c0 += 4) {
    const int cb = c0 + (half << 1);
    v2f a, bw;
    a.x  = xb[(size_t)cb * HW_];        // two contiguous 64B segments / wave
    a.y  = xb[(size_t)(cb + 1) * HW_];
    bw.x = w[cb];                       // uniform broadcast loads
    bw.y = w[cb + 1];
    acc = __builtin_amdgcn_wmma_f32_16x16x4_f32(
        /*neg_a=*/false, a, /*neg_b=*/false, bw,
        /*c_mod=*/(short)0, acc, /*reuse_a=*/false, /*reuse_b=*/false);
  }

  if (m == 0) {                         // lanes 0 (M=0..7) and 16 (M=8..15)
    float* zp = z + pbase + (half << 3);
    zp[0] = acc[0]; zp[1] = acc[1]; zp[2] = acc[2]; zp[3] = acc[3];
    zp[4] = acc[4]; zp[5] = acc[5]; zp[6] = acc[6]; zp[7] = acc[7];
  }
}

// ---------------------------------------------------------------------------
// Kernel 2: out[b][ho][wo] = relu( bilinear4(z)[b][ho][wo] + bias )
// Align-corners index math mirrors the reference (f32 pos, floor, clip,
// row-lerp then col-lerp). z (491 KB) is L2-resident; output stores coalesced.
// ---------------------------------------------------------------------------
__global__ __launch_bounds__(256) void
upsample_lerp_bias_relu_kernel(const float* __restrict__ z,
                               const float* __restrict__ bias,
                               float* __restrict__ out) {
  const int idx = blockIdx.x * 256 + threadIdx.x;   // grid exactly covers NOUT_
  const int wo  = idx % WO_;
  const int t   = idx / WO_;
  const int ho  = t % HO_;
  const int b   = t / HO_;

  const float RY = (float)((double)(H_ - 1) / (double)(HO_ - 1));  // 95/383
  const float RX = (float)((double)(W_ - 1) / (double)(WO_ - 1));  // 319/1279

  const float fy = (float)ho * RY;
  int i0 = (int)fy; if (i0 > H_ - 2) i0 = H_ - 2;
  const float wy = fy - (float)i0;

  const float fx = (float)wo * RX;
  int j0 = (int)fx; if (j0 > W_ - 2) j0 = W_ - 2;
  const float wx = fx - (float)j0;

  const float* zb = z + b * HW_ + i0 * W_ + j0;
  const float v00 = zb[0];
  const float v01 = zb[1];
  const float v10 = zb[W_];
  const float v11 = zb[W_ + 1];

  const float r0  = v00 * (1.0f - wy) + v10 * wy;   // rows lerp (y) first
  const float r1  = v01 * (1.0f - wy) + v11 * wy;
  const float val = r0 * (1.0f - wx) + r1 * wx + bias[0];

  out[idx] = val > 0.0f ? val : 0.0f;
}

extern "C" void kernel_launch(void* const* d_in, const int* in_sizes, int n_in,
                              void* d_out, int out_size, void* d_ws, size_t ws_size,
                              hipStream_t stream) {
  (void)in_sizes; (void)n_in; (void)out_size; (void)ws_size;

  const float* x    = (const float*)d_in[0];   // (4,64,96,320)
  const float* wgt  = (const float*)d_in[1];   // 64 floats
  const float* bias = (const float*)d_in[2];   // 1 float
  float*       out  = (float*)d_out;           // (4,1,384,1280)
  float*       z    = (float*)d_ws;            // 122880 floats scratch

  // 7680 waves * 16 positions = 122880 = NZ_, no tail.
  upsample_reduce_c_kernel<<<NZ_ / 16 / 8, 256, 0, stream>>>(x, wgt, z);
  // 7680 blocks * 256 threads = 1966080 = NOUT_, no tail.
  upsample_lerp_bias_relu_kernel<<<NOUT_ / 256, 256, 0, stream>>>(z, bias, out);
}